// Aggregation_61847529062503
// MI455X (gfx1250) — compile-verified
//
#include <hip/hip_runtime.h>
#include <cstdint>

// ---------------------------------------------------------------------------
// Segment-mean aggregation (MeanAggregation) for MI455X / gfx1250.
//   H_v:   [N_NODES, 512] f32 (rows of each graph are contiguous)
//   sizes: [N_GRAPHS] i32
//   out:   [N_GRAPHS, 512] f32  = per-graph mean (zeros for empty graphs)
// Memory-bound (~1.1 GB traffic => ~47us floor @ 23.3 TB/s). Uses CDNA5
// async global->LDS copies (ASYNCcnt) with double buffering.
// ---------------------------------------------------------------------------

#define D_DIM 512

constexpr int ROWS_PER_CHUNK  = 8;
constexpr int CHUNK_FLOATS    = ROWS_PER_CHUNK * D_DIM;   // 4096 floats = 16 KB
constexpr int CHUNK_VEC4      = CHUNK_FLOATS / 4;         // 1024 float4 slots
constexpr int THREADS         = 256;
constexpr int VEC4_PER_THREAD = CHUNK_VEC4 / THREADS;     // 4 async b128 / thread / chunk

// ---------------------------------------------------------------------------
// Kernel 1: exclusive prefix sum of sizes -> row offsets (single block).
// 1024 threads (32 wave32s), 16 elements per thread.
// ---------------------------------------------------------------------------
__global__ __launch_bounds__(1024) void seg_offsets_kernel(
    const int* __restrict__ sizes, int* __restrict__ offsets, int n) {
  const int T = 1024;
  const int tid = threadIdx.x;
  int per = (n + T - 1) / T;      // 16 for n = 16384
  if (per > 16) per = 16;         // local[] bound
  const int base = tid * per;

  int local[16];
  int sum = 0;
#pragma unroll
  for (int i = 0; i < 16; ++i) {
    int v = 0;
    if (i < per && (base + i) < n) v = sizes[base + i];
    local[i] = v;
    sum += v;
  }

  const int lane = tid & 31;
  const int wave = tid >> 5;

  // inclusive wave scan of per-thread totals (wave32)
  int inc = sum;
#pragma unroll
  for (int d = 1; d < 32; d <<= 1) {
    int u = __shfl_up(inc, d, 32);
    if (lane >= d) inc += u;
  }

  __shared__ int wsum[32];
  if (lane == 31) wsum[wave] = inc;
  __syncthreads();

  if (wave == 0) {
    int w = wsum[lane];
#pragma unroll
    for (int d = 1; d < 32; d <<= 1) {
      int u = __shfl_up(w, d, 32);
      if (lane >= d) w += u;
    }
    wsum[lane] = w;   // inclusive scan of wave totals
  }
  __syncthreads();

  int excl = ((wave > 0) ? wsum[wave - 1] : 0) + (inc - sum);
  for (int i = 0; i < per; ++i) {
    if ((base + i) < n) offsets[base + i] = excl;
    excl += local[i];
  }
}

// ---------------------------------------------------------------------------
// Kernel 2: one 256-thread block per graph. Double-buffered async staging of
// 8-row (16 KB) chunks into LDS, then per-thread accumulation of 2 columns.
// ---------------------------------------------------------------------------
__global__ __launch_bounds__(THREADS) void seg_mean_kernel(
    const float* __restrict__ H, const int* __restrict__ sizes,
    const int* __restrict__ offsets, float* __restrict__ out) {
  __shared__ float buf[2][CHUNK_FLOATS];   // 2 x 16 KB

  const int g = blockIdx.x;
  const int t = threadIdx.x;
  const int cnt = sizes[g];
  const int start = offsets[g];
  const float* __restrict__ src = H + (size_t)start * D_DIM;

  float accx = 0.0f, accy = 0.0f;

  if (cnt > 0) {
    const int tot4 = cnt * (D_DIM / 4);                        // total float4s in slab
    const int nchunks = (cnt + ROWS_PER_CHUNK - 1) / ROWS_PER_CHUNK;
    const uint32_t lds0 = (uint32_t)(uintptr_t)&buf[0][0];

    // Issue chunk c into buffer bi: exactly VEC4_PER_THREAD async b128 per
    // thread (unconditional -> per-wave ASYNCcnt increments are uniform).
    // Out-of-range slots are clamped in-bounds; their LDS rows are never read.
    auto issue_chunk = [&](int c, int bi) {
#pragma unroll
      for (int i = 0; i < VEC4_PER_THREAD; ++i) {
        const int slot = t + i * THREADS;            // 0..1023 within chunk
        int g4 = c * CHUNK_VEC4 + slot;
        if (g4 > tot4 - 1) g4 = tot4 - 1;            // clamp (cnt>0 => tot4>=128)
        const float* gp = src + (size_t)g4 * 4;
        const uint32_t lds_addr =
            lds0 + (uint32_t)bi * (CHUNK_FLOATS * 4) + (uint32_t)slot * 16;
        asm volatile("global_load_async_to_lds_b128 %0, %1, off"
                     :
                     : "v"(lds_addr), "v"(gp)
                     : "memory");
      }
    };

    issue_chunk(0, 0);
    for (int c = 0; c < nchunks; ++c) {
      const int bi = c & 1;
      const bool more = (c + 1) < nchunks;
      if (more) issue_chunk(c + 1, bi ^ 1);          // keep next chunk in flight

      // Wait for chunk c only (4 async ops per chunk per wave, in-order done).
      if (more) {
        asm volatile("s_wait_asynccnt 4" ::: "memory");
      } else {
        asm volatile("s_wait_asynccnt 0" ::: "memory");
      }
      __syncthreads();   // publish LDS writes from all waves' async ops

      const int rows = min(ROWS_PER_CHUNK, cnt - c * ROWS_PER_CHUNK);
      const float* bp = &buf[bi][2 * t];
#pragma unroll 8
      for (int r = 0; r < rows; ++r) {
        const float2 v = *(const float2*)(bp + r * D_DIM);
        accx += v.x;
        accy += v.y;
      }
      __syncthreads();   // buffer bi free for reuse two iterations later
    }
  }

  const float inv = 1.0f / (float)((cnt > 0) ? cnt : 1);
  float2 o;
  o.x = accx * inv;
  o.y = accy * inv;
  *(float2*)&out[(size_t)g * D_DIM + 2 * t] = o;
}

// ---------------------------------------------------------------------------
// Entry point
// ---------------------------------------------------------------------------
extern "C" void kernel_launch(void* const* d_in, const int* in_sizes, int n_in,
                              void* d_out, int out_size, void* d_ws, size_t ws_size,
                              hipStream_t stream) {
  const float* H     = (const float*)d_in[0];   // [N_NODES, 512] f32
  const int*   sizes = (const int*)d_in[1];     // [N_GRAPHS] i32
  const int b = in_sizes[1];                    // number of graphs (16384)

  int* offsets = (int*)d_ws;                    // b ints of scratch

  seg_offsets_kernel<<<1, 1024, 0, stream>>>(sizes, offsets, b);
  seg_mean_kernel<<<b, THREADS, 0, stream>>>(H, sizes, offsets, (float*)d_out);
}